// ViewTransformer_20907900797225
// MI455X (gfx1250) — compile-verified
//
#include <hip/hip_runtime.h>

typedef float v2f __attribute__((ext_vector_type(2)));
typedef float v8f __attribute__((ext_vector_type(8)));

#define B_ 2
#define N_ 6
#define C_ 128
#define H_ 16
#define W_ 44
#define D_ 64
#define BEV 256
#define NCELL (BEV * BEV)
#define CNT_OFF 256  // float offset of count buffer inside d_ws

// ---------------------------------------------------------------------------
// Kernel 0: zero the output accumulator and the count buffer (d_out / d_ws are
// poisoned by the harness, so we must clear before accumulating).
// ---------------------------------------------------------------------------
__global__ void bev_zero_kernel(float* __restrict__ out, float* __restrict__ cnt) {
    size_t stride = (size_t)gridDim.x * blockDim.x;
    size_t i = (size_t)blockIdx.x * blockDim.x + threadIdx.x;
    const size_t n_out = (size_t)B_ * C_ * NCELL;
    for (size_t k = i; k < n_out; k += stride) out[k] = 0.0f;
    const size_t n_cnt = (size_t)B_ * NCELL;
    for (size_t k = i; k < n_cnt; k += stride) cnt[k] = 0.0f;
}

// ---------------------------------------------------------------------------
// Kernel 1: per-(b,n) geometry. Scale K by [img_w/W, img_h/H, 1], invert 3x3,
// fold into M = R * K^-1 and translation t. 12 cameras total.
// ---------------------------------------------------------------------------
__global__ void bev_geom_kernel(const float* __restrict__ intr,
                                const float* __restrict__ extr,
                                const int* __restrict__ img_h,
                                const int* __restrict__ img_w,
                                float* __restrict__ geom) {
    int id = threadIdx.x;
    if (id >= B_ * N_) return;
    float sx = (float)img_w[0] / (float)W_;  // rs[0] = img_w / W
    float sy = (float)img_h[0] / (float)H_;  // rs[1] = img_h / H
    const float* K = intr + id * 9;
    float a = K[0] * sx, b = K[1] * sx, c = K[2] * sx;
    float d = K[3] * sy, e = K[4] * sy, f = K[5] * sy;
    float g = K[6],      h = K[7],      i9 = K[8];
    float det = a * (e * i9 - f * h) - b * (d * i9 - f * g) + c * (d * h - e * g);
    float inv = 1.0f / det;
    float Ki[9];
    Ki[0] =  (e * i9 - f * h) * inv; Ki[1] = -(b * i9 - c * h) * inv; Ki[2] =  (b * f - c * e) * inv;
    Ki[3] = -(d * i9 - f * g) * inv; Ki[4] =  (a * i9 - c * g) * inv; Ki[5] = -(a * f - c * d) * inv;
    Ki[6] =  (d * h - e * g) * inv;  Ki[7] = -(a * h - b * g) * inv;  Ki[8] =  (a * e - b * d) * inv;
    const float* E = extr + id * 16;  // 4x4 row-major
    float* gp = geom + id * 12;
    for (int r = 0; r < 3; ++r)
        for (int cc = 0; cc < 3; ++cc)
            gp[r * 3 + cc] = E[r * 4 + 0] * Ki[0 * 3 + cc] +
                             E[r * 4 + 1] * Ki[1 * 3 + cc] +
                             E[r * 4 + 2] * Ki[2 * 3 + cc];
    gp[9]  = E[3];
    gp[10] = E[7];
    gp[11] = E[11];
}

// ---------------------------------------------------------------------------
// Kernel 2: one 256-thread workgroup (8 wave32) per pixel (b,n,h,w).
//  - stage depth[64], feat[128] and geometry in LDS
//  - threads 0..63 compute BEV cell index per depth bin + count atomics
//  - each wave owns a 16-channel column; 4x V_WMMA_F32_16X16X4_F32 compute the
//    depth x feat outer product (depth in A's K=0 column, feat in B's K=0 row)
//  - indices for the d-tile are preloaded to registers BEFORE the WMMA so the
//    LDS latency hides under the matrix op and the atomic chain never waits
//    on dscnt; scatter uses native f32 atomics per the documented C/D layout
// ---------------------------------------------------------------------------
__global__ void bev_splat_kernel(const float* __restrict__ feat,
                                 const float* __restrict__ depth,
                                 const float* __restrict__ geom,
                                 float* __restrict__ out,
                                 float* __restrict__ cnt) {
    __shared__ float sM[12];
    __shared__ float sdep[D_];
    __shared__ float sfeat[C_];
    __shared__ int   sidx[D_];

    int pix = blockIdx.x;
    int w = pix % W_;
    int h = (pix / W_) % H_;
    int n = (pix / (W_ * H_)) % N_;
    int b = pix / (W_ * H_ * N_);
    int bn = b * N_ + n;
    int tid = threadIdx.x;
    int hw = h * W_ + w;

    if (tid < 12)  sM[tid]    = geom[bn * 12 + tid];
    if (tid < D_)  sdep[tid]  = depth[((size_t)(bn * D_ + tid)) * (H_ * W_) + hw];
    if (tid < C_)  sfeat[tid] = feat[((size_t)(bn * C_ + tid)) * (H_ * W_) + hw];
    __syncthreads();

    if (tid < D_) {
        float dd = 1.0f + (float)tid * (59.0f / 63.0f);  // linspace(1,60,64)
        float u = (float)w, v = (float)h;
        float px = (sM[0] * u + sM[1] * v + sM[2]) * dd + sM[9];
        float py = (sM[3] * u + sM[4] * v + sM[5]) * dd + sM[10];
        float pz = (sM[6] * u + sM[7] * v + sM[8]) * dd + sM[11];
        int xi = (int)((px + 51.2f) / 0.4f);  // trunc-toward-zero == astype(int32)
        int yi = (int)((py + 51.2f) / 0.4f);
        bool valid = (xi >= 0) && (xi < BEV) && (yi >= 0) && (yi < BEV) &&
                     (pz >= -5.0f) && (pz <= 3.0f);
        int idx = valid ? (yi * BEV + xi) : -1;
        sidx[tid] = idx;
        if (valid) {
            __hip_atomic_fetch_add(&cnt[b * NCELL + idx], 1.0f,
                                   __ATOMIC_RELAXED, __HIP_MEMORY_SCOPE_AGENT);
        }
    }
    __syncthreads();

    int wv   = tid >> 5;        // wave id 0..7 -> channel tile
    int lane = tid & 31;
    int c0   = wv * 16;
    int lm   = lane & 15;
    bool lo  = lane < 16;

    // B operand: feat[c0+n] in the K=0 row (VGPR0 lanes 0..15), zero elsewhere.
    float fv = sfeat[c0 + lm];
    v2f Bm;
    Bm.x = lo ? fv : 0.0f;
    Bm.y = 0.0f;

    float* obase = out + (((size_t)(b * C_ + c0 + lm)) << 16);  // 65536 cells/chan
    int mbase = lo ? 0 : 8;  // C/D layout: VGPR r -> M=r (lanes 0-15) / M=r+8

#pragma unroll
    for (int dt = 0; dt < 4; ++dt) {
        int d0 = dt * 16;
        // A operand: depth[d0+m] in the K=0 column (VGPR0 lanes 0..15).
        float dv = sdep[d0 + lm];
        v2f Am;
        Am.x = lo ? dv : 0.0f;
        Am.y = 0.0f;

        // Preload this d-tile's cell indices into registers; the ds_load
        // latency overlaps the WMMA below instead of stalling each atomic.
        int idxr[8];
#pragma unroll
        for (int r = 0; r < 8; ++r) idxr[r] = sidx[d0 + mbase + r];

        v8f acc = {};
        // D[m][n] = depth[d0+m] * feat[c0+n], exact fp32 products
        acc = __builtin_amdgcn_wmma_f32_16x16x4_f32(
            /*neg_a=*/false, Am, /*neg_b=*/false, Bm,
            /*c_mod=*/(short)0, acc, /*reuse_a=*/false, /*reuse_b=*/false);

#pragma unroll
        for (int r = 0; r < 8; ++r) {
            int idx = idxr[r];
            if (idx >= 0) {
                __hip_atomic_fetch_add(obase + idx, acc[r],
                                       __ATOMIC_RELAXED, __HIP_MEMORY_SCOPE_AGENT);
            }
        }
    }
}

// ---------------------------------------------------------------------------
// Kernel 3: bev / (cnt + 1e-5), float4-vectorized.
// ---------------------------------------------------------------------------
__global__ void bev_norm_kernel(float* __restrict__ out, const float* __restrict__ cnt) {
    size_t total4 = (size_t)B_ * C_ * NCELL / 4;
    size_t i4 = (size_t)blockIdx.x * blockDim.x + threadIdx.x;
    if (i4 >= total4) return;
    size_t i = i4 * 4;
    int b    = (int)(i >> 23);            // C_*NCELL = 2^23
    int cell = (int)(i & (NCELL - 1));    // 4-aligned, no wrap within float4
    float4 v = reinterpret_cast<float4*>(out)[i4];
    float4 c = reinterpret_cast<const float4*>(cnt)[(size_t)(b * NCELL + cell) >> 2];
    v.x /= (c.x + 1e-5f);
    v.y /= (c.y + 1e-5f);
    v.z /= (c.z + 1e-5f);
    v.w /= (c.w + 1e-5f);
    reinterpret_cast<float4*>(out)[i4] = v;
}

extern "C" void kernel_launch(void* const* d_in, const int* in_sizes, int n_in,
                              void* d_out, int out_size, void* d_ws, size_t ws_size,
                              hipStream_t stream) {
    const float* feat  = (const float*)d_in[0];
    const float* depth = (const float*)d_in[1];
    const float* intr  = (const float*)d_in[2];
    const float* extr  = (const float*)d_in[3];
    const int*   img_h = (const int*)d_in[4];
    const int*   img_w = (const int*)d_in[5];
    float* out  = (float*)d_out;
    float* geom = (float*)d_ws;
    float* cnt  = (float*)d_ws + CNT_OFF;

    bev_zero_kernel<<<2048, 256, 0, stream>>>(out, cnt);
    bev_geom_kernel<<<1, 32, 0, stream>>>(intr, extr, img_h, img_w, geom);
    bev_splat_kernel<<<B_ * N_ * H_ * W_, 256, 0, stream>>>(feat, depth, geom, out, cnt);
    size_t total4 = (size_t)B_ * C_ * NCELL / 4;
    bev_norm_kernel<<<(unsigned)((total4 + 255) / 256), 256, 0, stream>>>(out, cnt);
}